// SimpleNet_22454089023572
// MI455X (gfx1250) — compile-verified
//
#include <hip/hip_runtime.h>
#include <math.h>

#define Bx 32
#define Hh 96
#define Ww 96
#define HW (Hh*Ww)
#define NEGV (-1e9f)

typedef __attribute__((ext_vector_type(2))) float v2f;
typedef __attribute__((ext_vector_type(8))) float v8f;

// ---------------- float atomic min/max via CAS ----------------
__device__ __forceinline__ void atomicMinF(float* addr, float val) {
    unsigned int* ai = (unsigned int*)addr;
    unsigned int old = *ai;
    while (true) {
        float f = __uint_as_float(old);
        if (f <= val) break;
        unsigned int assumed = old;
        old = atomicCAS(ai, assumed, __float_as_uint(val));
        if (old == assumed) break;
    }
}
__device__ __forceinline__ void atomicMaxF(float* addr, float val) {
    unsigned int* ai = (unsigned int*)addr;
    unsigned int old = *ai;
    while (true) {
        float f = __uint_as_float(old);
        if (f >= val) break;
        unsigned int assumed = old;
        old = atomicCAS(ai, assumed, __float_as_uint(val));
        if (old == assumed) break;
    }
}

// ---------------- kernel 0: init outputs, minmax cells, padded weights ----------------
__global__ void k_init(float* __restrict__ out, float* __restrict__ mmx,
                       const float* __restrict__ Wd, float* __restrict__ wpad) {
    int i = blockIdx.x * 256 + threadIdx.x;
    if (i < 3 * Bx * 1000) out[i] = 0.f;
    if (i < 16 * 48) {                       // Wpad[16][48]: rows>=4 and cols>=45 are zero
        int m = i / 48;
        int f = i - m * 48;
        wpad[i] = (m < 4 && f < 45) ? Wd[m * 45 + f] : 0.f;
    }
    if (i == 0) {
        mmx[0] = 3.0e38f;  mmx[1] = 3.0e38f;   // min ch0, ch1
        mmx[2] = -3.0e38f; mmx[3] = -3.0e38f;  // max ch0, ch1
    }
}

// ---------------- kernel 1: le = avgpool3x3(mf)+mf, global min/max ----------------
__global__ void k_le(const float* __restrict__ mf, float* __restrict__ le,
                     float* __restrict__ mmx) {
    int gid = blockIdx.x * 256 + threadIdx.x;     // pixel id over B*HW
    int b = gid / HW;
    int off = gid - b * HW;
    int y = off / Ww;
    int x = off - y * Ww;
    float v[2];
    #pragma unroll
    for (int c = 0; c < 2; c++) {
        const float* p = mf + ((size_t)b * 2 + c) * HW;
        float s = 0.f;
        #pragma unroll
        for (int dy = -1; dy <= 1; dy++) {
            int yy = y + dy;
            if (yy < 0 || yy >= Hh) continue;
            #pragma unroll
            for (int dx = -1; dx <= 1; dx++) {
                int xx = x + dx;
                if (xx < 0 || xx >= Ww) continue;
                s += p[yy * Ww + xx];
            }
        }
        float l = s * (1.f / 9.f) + p[off];
        le[((size_t)b * 2 + c) * HW + off] = l;
        v[c] = l;
    }
    __shared__ float red[256];
    #pragma unroll
    for (int q = 0; q < 4; q++) {   // min0, min1, max0, max1
        bool ismin = (q < 2);
        float val = (q & 1) ? v[1] : v[0];
        red[threadIdx.x] = val;
        __syncthreads();
        for (int s = 128; s > 0; s >>= 1) {
            if ((int)threadIdx.x < s) {
                float a = red[threadIdx.x], bb = red[threadIdx.x + s];
                red[threadIdx.x] = ismin ? fminf(a, bb) : fmaxf(a, bb);
            }
            __syncthreads();
        }
        if (threadIdx.x == 0) {
            if (ismin) atomicMinF(&mmx[q], red[0]);
            else       atomicMaxF(&mmx[q], red[0]);
        }
        __syncthreads();
    }
}

// ---------------- kernel 2a: stage in5 = [norm(le0), norm(le1), uf0..2] ----------------
__global__ void k_stage(const float* __restrict__ le, const float* __restrict__ uf,
                        const float* __restrict__ mmx, float* __restrict__ in5) {
    int gid = blockIdx.x * 256 + threadIdx.x;
    int b = gid / HW;
    int off = gid - b * HW;
    float mn0 = mmx[0], mn1 = mmx[1], mx0 = mmx[2], mx1 = mmx[3];
    float s0 = 2.f / (mx0 - mn0), t0 = -2.f * mn0 / (mx0 - mn0) - 1.f;
    float s1 = 2.f / (mx1 - mn1), t1 = -2.f * mn1 / (mx1 - mn1) - 1.f;
    float* dst = in5 + (size_t)b * 5 * HW + off;
    dst[0 * HW] = s0 * le[((size_t)b * 2 + 0) * HW + off] + t0;
    dst[1 * HW] = s1 * le[((size_t)b * 2 + 1) * HW + off] + t1;
    dst[2 * HW] = uf[((size_t)b * 3 + 0) * HW + off];
    dst[3 * HW] = uf[((size_t)b * 3 + 1) * HW + off];
    dst[4 * HW] = uf[((size_t)b * 3 + 2) * HW + off];
}

// ---------------- kernel 2b: conv3x3 (5->4) via WMMA f32 16x16x4 ----------------
__device__ __forceinline__ float lrelu(float v) { return v > 0.f ? v : 0.01f * v; }

__global__ void k_conv(const float* __restrict__ in5, const float* __restrict__ wpad,
                       const float* __restrict__ bd, float4* __restrict__ dfout) {
    int lane = threadIdx.x & 31;
    int tile = blockIdx.x * 8 + (threadIdx.x >> 5);   // 16 pixels per wave tile
    int b = tile / (Hh * 6);
    int r = tile - b * (Hh * 6);
    int y = r / 6;
    int x0 = (r - y * 6) * 16;

    int m = lane & 15;              // A row (output channel; rows >=4 zero via wpad)
    int n = lane & 15;              // B column (pixel within tile)
    int kb = (lane < 16) ? 0 : 2;   // K-half held by this lane (ISA 16x4 striping)
    int px = x0 + n;

    const float* base = in5 + (size_t)b * 5 * HW;
    const float* wrow = wpad + m * 48;

    // Phase 1: gather all operands unconditionally (clamped addr + select),
    // so the compiler can batch the loads into one burst.
    float avals[24], bvals[24];
    #pragma unroll
    for (int kc = 0; kc < 12; kc++) {
        #pragma unroll
        for (int e = 0; e < 2; e++) {
            int flat = kc * 4 + kb + e;          // < 48
            int f = flat < 45 ? flat : 0;
            int i = f / 9;
            int rr = f - i * 9;
            int ky = rr / 3;
            int kx = rr - ky * 3;
            int yy = y + ky - 1;
            int xx = px + kx - 1;
            bool ok = (flat < 45) & (yy >= 0) & (yy < Hh) & (xx >= 0) & (xx < Ww);
            int yyc = min(max(yy, 0), Hh - 1);
            int xxc = min(max(xx, 0), Ww - 1);
            float v = base[i * HW + yyc * Ww + xxc];   // always-valid load
            bvals[kc * 2 + e] = ok ? v : 0.f;
            avals[kc * 2 + e] = wrow[flat];            // always-valid load
        }
    }

    // Phase 2: 12 back-to-back WMMAs on the accumulator chain (EXEC all ones).
    v8f acc = {};
    #pragma unroll
    for (int kc = 0; kc < 12; kc++) {
        v2f av, bv;
        av.x = avals[kc * 2];
        av.y = avals[kc * 2 + 1];
        bv.x = bvals[kc * 2];
        bv.y = bvals[kc * 2 + 1];
        acc = __builtin_amdgcn_wmma_f32_16x16x4_f32(false, av, false, bv,
                                                    (short)0, acc, false, false);
    }

    if (lane < 16) {   // rows 0..3 of D live in lanes 0-15, VGPRs 0..3
        float4 o;
        o.x = lrelu(acc[0] + bd[0]);
        o.y = lrelu(acc[1] + bd[1]);
        o.z = lrelu(acc[2] + bd[2]);
        o.w = lrelu(acc[3] + bd[3]);
        dfout[((size_t)b * Hh + y) * Ww + px] = o;
    }
}

// ---------------- masked categorical head ----------------
template <int K, int C>
__device__ __forceinline__ void mcat(const float* __restrict__ W, const float* __restrict__ bias,
                                     const float* x, unsigned int mask,
                                     float& sel, int& arg, float& ent) {
    float l[K];
    #pragma unroll
    for (int k = 0; k < K; k++) {
        float a = bias[k];
        #pragma unroll
        for (int c = 0; c < C; c++) a += W[k * C + c] * x[c];
        l[k] = ((mask >> k) & 1u) ? a : NEGV;
    }
    float mx = l[0]; int am = 0;
    #pragma unroll
    for (int k = 1; k < K; k++) if (l[k] > mx) { mx = l[k]; am = k; }
    float s = 0.f;
    #pragma unroll
    for (int k = 0; k < K; k++) s += expf(l[k] - mx);
    float lse = mx + logf(s);
    sel = l[am] - lse;
    float e = 0.f;
    #pragma unroll
    for (int k = 0; k < K; k++)
        if ((mask >> k) & 1u) { float lp = l[k] - lse; e -= expf(lp) * lp; }
    ent = e;
    arg = am;
}

// ---------------- kernel 3: fused per-pixel heads + scatter ----------------
__global__ void k_main(const float* __restrict__ mf, const float* __restrict__ ff,
                       const float* __restrict__ uf, const int* __restrict__ loc,
                       const unsigned char* __restrict__ vam, const unsigned char* __restrict__ vat,
                       const unsigned char* __restrict__ vap, const unsigned char* __restrict__ vad,
                       const unsigned char* __restrict__ vasd, const unsigned char* __restrict__ varc,
                       const unsigned char* __restrict__ vadn, const unsigned char* __restrict__ vafa,
                       const float* Wf, const float* bf, const float* Wc, const float* bc,
                       const float* Wt, const float* bt, const float* Wmd, const float* bmd,
                       const float* Wtd, const float* btd, const float* Wtr, const float* btr,
                       const float* Wta, const float* bta, const float* Wpr, const float* bpr,
                       const float* Wpa, const float* bpa, const float* Wra, const float* bra,
                       const float* Wdr, const float* bdr,
                       const float4* __restrict__ dfbuf, float* __restrict__ out) {
    int gid = blockIdx.x * 256 + threadIdx.x;
    int b = gid / HW;
    int off = gid - b * HW;

    // ---- mask reductions ----
    unsigned mvbits = 0;
    {
        const unsigned char* pm = vam + (size_t)b * 10 * HW + off;
        #pragma unroll
        for (int d = 0; d < 5; d++) {
            unsigned char m = pm[(size_t)(d * 2) * HW] | pm[(size_t)(d * 2 + 1) * HW];
            mvbits |= (m ? 1u : 0u) << d;
        }
    }
    unsigned tvbits = 0;   // bit d*5+r : any over amount
    {
        const unsigned char* pt = vat + (size_t)b * 250 * HW + off;
        for (int d = 0; d < 5; d++)
            for (int r = 0; r < 5; r++) {
                const unsigned char* q = pt + (size_t)((d * 5 + r) * 10) * HW;
                bool f = false;
                for (int a = 0; a < 10; a++) {
                    if (q[(size_t)a * HW]) { f = true; break; }
                }
                tvbits |= (f ? 1u : 0u) << (d * 5 + r);
            }
    }
    unsigned trdbits = 0;
    #pragma unroll
    for (int d = 0; d < 5; d++) trdbits |= (((tvbits >> (d * 5)) & 31u) ? 1u : 0u) << d;
    unsigned pkbits = 0;
    {
        const unsigned char* pp = vap + (size_t)b * 50 * HW + off;
        for (int r = 0; r < 5; r++) {
            const unsigned char* q = pp + (size_t)(r * 10) * HW;
            bool f = false;
            for (int a = 0; a < 10; a++) {
                if (q[(size_t)a * HW]) { f = true; break; }
            }
            pkbits |= (f ? 1u : 0u) << r;
        }
    }
    unsigned digbits = (vad[(size_t)b * 2 * HW + off] ? 1u : 0u) |
                       (vad[((size_t)b * 2 + 1) * HW + off] ? 2u : 0u);
    bool sd = vasd[(size_t)b * 2 * HW + off] | vasd[((size_t)b * 2 + 1) * HW + off];
    bool rc = varc[(size_t)b * 2 * HW + off] | varc[((size_t)b * 2 + 1) * HW + off];
    bool dn = vadn[(size_t)b * HW + off] != 0;
    unsigned fabits = 0;
    #pragma unroll
    for (int k = 0; k < 4; k++) fabits |= (vafa[((size_t)b * 4 + k) * HW + off] ? 1u : 0u) << k;

    unsigned uat = (mvbits ? 1u : 0u) | ((tvbits ? 1u : 0u) << 1) | ((pkbits ? 1u : 0u) << 2) |
                   ((digbits ? 1u : 0u) << 3) | ((sd ? 1u : 0u) << 4) | ((rc ? 1u : 0u) << 5) |
                   ((dn ? 1u : 0u) << 6);
    bool unit_mask = uat != 0;
    bool fmask = fabits != 0;

    // ---- features ----
    float vf[5];
    vf[0] = mf[((size_t)b * 2 + 0) * HW + off];
    vf[1] = mf[((size_t)b * 2 + 1) * HW + off];
    vf[2] = uf[((size_t)b * 3 + 0) * HW + off];
    vf[3] = uf[((size_t)b * 3 + 1) * HW + off];
    vf[4] = uf[((size_t)b * 3 + 2) * HW + off];
    float f6[6];
    #pragma unroll
    for (int k = 0; k < 6; k++) f6[k] = ff[((size_t)b * 6 + k) * HW + off];
    float4 d4 = dfbuf[gid];
    float dfa[4] = { d4.x, d4.y, d4.z, d4.w };

    float critic = bc[0];
    #pragma unroll
    for (int c = 0; c < 5; c++) critic += Wc[c] * vf[c];

    // ---- heads ----
    float t_lp, t_en; int t;
    mcat<7, 5>(Wt, bt, vf, uat, t_lp, t, t_en);
    float du0, du1; int f_act;
    mcat<4, 6>(Wf, bf, f6, fabits, du0, f_act, du1);

    float lp_mvd, en_mvd; int mv_dir;
    mcat<5, 4>(Wmd, bmd, dfa, mvbits, lp_mvd, mv_dir, en_mvd);
    float lp_trd, en_trd; int tr_dir;
    mcat<5, 4>(Wtd, btd, dfa, trdbits, lp_trd, tr_dir, en_trd);
    unsigned trrbits = (tvbits >> (tr_dir * 5)) & 31u;
    float lp_trr, en_trr; int tr_res;
    mcat<5, 4>(Wtr, btr, dfa, trrbits, lp_trr, tr_res, en_trr);
    float lp_tra, en_tra; int tr_amt;
    mcat<10, 4>(Wta, bta, dfa, 0x3FFu, lp_tra, tr_amt, en_tra);
    float lp_pkr, en_pkr; int pk_res;
    mcat<5, 4>(Wpr, bpr, dfa, pkbits, lp_pkr, pk_res, en_pkr);
    float lp_pka, en_pka; int pk_amt;
    mcat<10, 4>(Wpa, bpa, dfa, 0x3FFu, lp_pka, pk_amt, en_pka);
    float lp_rca, en_rca; int rc_amt;
    mcat<10, 4>(Wra, bra, dfa, 0x3FFu, lp_rca, rc_amt, en_rca);
    float lp_dgr, en_dgr; int dg_rep;
    mcat<2, 4>(Wdr, bdr, dfa, digbits, lp_dgr, dg_rep, en_dgr);

    // ---- per-type composition ----
    float lp_t[7] = { lp_mvd, lp_trd + lp_trr + lp_tra, lp_pkr + lp_pka, lp_dgr, 0.f, lp_rca, 0.f };
    float en_t[7] = { en_mvd, en_trd + en_trr + en_tra, en_pkr + en_pka, en_dgr, 0.f, en_rca, 0.f };
    float total_lp = t_lp + lp_t[t];
    float total_en = t_en + en_t[t];
    int direction = (t == 0) ? mv_dir : (t == 1) ? tr_dir : 0;
    int resource  = (t == 1) ? tr_res : (t == 2) ? pk_res : 0;
    int amount    = (t == 1) ? tr_amt : (t == 2) ? pk_amt : (t == 5) ? rc_amt : 0;
    int repeat    = (t == 3) ? dg_rep : 0;

    // ---- outputs ----
    float* out_lp = out;
    float* out_cr = out + 32 * 1000;
    float* out_en = out + 64 * 1000;
    float* out_fa = out + 96 * 1000;
    float* out_ua = out_fa + (size_t)Bx * HW;

    out_fa[(size_t)b * HW + off] = fmask ? (float)f_act : 0.f;

    float um = unit_mask ? 1.f : 0.f;
    out_ua[((size_t)b * 6 + 0) * HW + off] = um * (float)t;
    out_ua[((size_t)b * 6 + 1) * HW + off] = um * (float)direction;
    out_ua[((size_t)b * 6 + 2) * HW + off] = um * (float)resource;
    out_ua[((size_t)b * 6 + 3) * HW + off] = um * (float)amount;
    out_ua[((size_t)b * 6 + 4) * HW + off] = um * (float)repeat;
    out_ua[((size_t)b * 6 + 5) * HW + off] = um;

    if (unit_mask) {
        int uid = loc[((size_t)b * 2 + 1) * HW + off] + 10;   // in [10,910)
        out_lp[b * 1000 + uid] = total_lp;
        out_cr[b * 1000 + uid] = critic;
        out_en[b * 1000 + uid] = total_en;
    }
}

// ---------------- launch ----------------
extern "C" void kernel_launch(void* const* d_in, const int* in_sizes, int n_in,
                              void* d_out, int out_size, void* d_ws, size_t ws_size,
                              hipStream_t stream) {
    const float* mf = (const float*)d_in[1];
    const float* ff = (const float*)d_in[2];
    const float* uf = (const float*)d_in[3];
    const int* loc = (const int*)d_in[4];
    const unsigned char* vam = (const unsigned char*)d_in[5];
    const unsigned char* vat = (const unsigned char*)d_in[6];
    const unsigned char* vap = (const unsigned char*)d_in[7];
    const unsigned char* vad = (const unsigned char*)d_in[8];
    const unsigned char* vasd = (const unsigned char*)d_in[9];
    const unsigned char* varc = (const unsigned char*)d_in[10];
    const unsigned char* vadn = (const unsigned char*)d_in[11];
    const unsigned char* vafa = (const unsigned char*)d_in[12];
    const float* Wf = (const float*)d_in[13];  const float* bf = (const float*)d_in[14];
    const float* Wc = (const float*)d_in[15];  const float* bc = (const float*)d_in[16];
    const float* Wd = (const float*)d_in[17];  const float* bd = (const float*)d_in[18];
    const float* Wt = (const float*)d_in[19];  const float* bt = (const float*)d_in[20];
    const float* Wmd = (const float*)d_in[21]; const float* bmd = (const float*)d_in[22];
    const float* Wtd = (const float*)d_in[23]; const float* btd = (const float*)d_in[24];
    const float* Wtr = (const float*)d_in[25]; const float* btr = (const float*)d_in[26];
    const float* Wta = (const float*)d_in[27]; const float* bta = (const float*)d_in[28];
    const float* Wpr = (const float*)d_in[29]; const float* bpr = (const float*)d_in[30];
    const float* Wpa = (const float*)d_in[31]; const float* bpa = (const float*)d_in[32];
    const float* Wra = (const float*)d_in[33]; const float* bra = (const float*)d_in[34];
    const float* Wdr = (const float*)d_in[35]; const float* bdr = (const float*)d_in[36];

    // workspace layout (d_ws is 256-aligned from hipMalloc)
    char* wsb = (char*)d_ws;
    float* mmx  = (float*)wsb;                                  // 4 floats @ 0
    float* wpad = (float*)(wsb + 256);                          // 768 floats (3 KB)
    float* le   = (float*)(wsb + 256 + 3072);                   // [B,2,H,W]  (2.36 MB)
    float* in5  = (float*)(wsb + 256 + 3072 + sizeof(float) * 2 * Bx * HW);      // [B,5,H,W] (5.9 MB)
    float4* df  = (float4*)(wsb + 256 + 3072 + sizeof(float) * 7 * Bx * HW);     // [B,H,W]x4 (4.7 MB)

    float* out = (float*)d_out;

    k_init<<<375, 256, 0, stream>>>(out, mmx, Wd, wpad);
    k_le<<<(Bx * HW) / 256, 256, 0, stream>>>(mf, le, mmx);
    k_stage<<<(Bx * HW) / 256, 256, 0, stream>>>(le, uf, mmx, in5);
    k_conv<<<(Bx * Hh * 6) / 8, 256, 0, stream>>>(in5, wpad, bd, df);
    k_main<<<(Bx * HW) / 256, 256, 0, stream>>>(
        mf, ff, uf, loc, vam, vat, vap, vad, vasd, varc, vadn, vafa,
        Wf, bf, Wc, bc, Wt, bt, Wmd, bmd, Wtd, btd, Wtr, btr, Wta, bta,
        Wpr, bpr, Wpa, bpa, Wra, bra, Wdr, bdr, df, out);
}